// SGDFilterDesign_18614388261610
// MI455X (gfx1250) — compile-verified
//
#include <hip/hip_runtime.h>

// ---------------------------------------------------------------------------
// Persistent single-WGP SGD filter-design kernel for MI455X (gfx1250).
//  * 1 block x 512 threads (16 wave32s), whole 1000-iteration loop on-chip.
//  * All optimizer state in LDS; global traffic = 2KB in + 384B out, once.
//  * Gradient reduction G[sec,coef] = sum_f J*w on V_WMMA_F32_16X16X4_F32,
//    full f32, layout-native (lane = section, K = frequency).
//  * Cross-lane log-magnitude sum: inline-asm batched butterfly -- 8
//    ds_bpermute_b32 in flight per s_wait_dscnt instead of 1, cutting the
//    per-iteration serialized LDS latency ~8x (this is the critical path).
//  * Hot transcendentals use raw v_log_f32 / v_exp_f32 / v_rcp_f32 in the
//    log2 domain (operands always normal; no denorm-guard code).
// ---------------------------------------------------------------------------

typedef float v2f __attribute__((ext_vector_type(2)));
typedef float v8f __attribute__((ext_vector_type(8)));

#define NF    512
#define NSEC  16
#define NIT   1000
#define LRATE 2e-5f
#define EPSV  1e-8f
#define PI_F  3.14159265358979323846f
#define K20L  8.6858896380650366f    /* 20 / ln(10)        */
#define K20L2 6.0205999132796239f    /* 20 * log10(2)      */
#define NSLOT 16                     /* freq slots per lane (8 groups x 2 K) */

// 8-wide batched butterfly step: 8 ds_bpermute_b32 issued back-to-back,
// one s_wait_dscnt draining all of them, then the adds.
#define BFLY8(ADDR, L, base)                                                  \
    do {                                                                      \
        float r0, r1, r2, r3, r4, r5, r6, r7;                                 \
        asm volatile(                                                         \
            "ds_bpermute_b32 %0, %8, %9\n\t"                                  \
            "ds_bpermute_b32 %1, %8, %10\n\t"                                 \
            "ds_bpermute_b32 %2, %8, %11\n\t"                                 \
            "ds_bpermute_b32 %3, %8, %12\n\t"                                 \
            "ds_bpermute_b32 %4, %8, %13\n\t"                                 \
            "ds_bpermute_b32 %5, %8, %14\n\t"                                 \
            "ds_bpermute_b32 %6, %8, %15\n\t"                                 \
            "ds_bpermute_b32 %7, %8, %16\n\t"                                 \
            "s_wait_dscnt 0x0"                                                \
            : "=&v"(r0), "=&v"(r1), "=&v"(r2), "=&v"(r3),                     \
              "=&v"(r4), "=&v"(r5), "=&v"(r6), "=&v"(r7)                      \
            : "v"(ADDR),                                                      \
              "v"((L)[(base) + 0]), "v"((L)[(base) + 1]),                     \
              "v"((L)[(base) + 2]), "v"((L)[(base) + 3]),                     \
              "v"((L)[(base) + 4]), "v"((L)[(base) + 5]),                     \
              "v"((L)[(base) + 6]), "v"((L)[(base) + 7]));                    \
        (L)[(base) + 0] += r0; (L)[(base) + 1] += r1;                         \
        (L)[(base) + 2] += r2; (L)[(base) + 3] += r3;                         \
        (L)[(base) + 4] += r4; (L)[(base) + 5] += r5;                         \
        (L)[(base) + 6] += r6; (L)[(base) + 7] += r7;                         \
    } while (0)

__global__ __launch_bounds__(512)
void sgd_filter_kernel(const float* __restrict__ target, float* __restrict__ out)
{
    __shared__ float s_tgt[NF];
    __shared__ float s_c1[NF], s_s1[NF], s_c2[NF], s_s2[NF];
    __shared__ float s_raw[NSEC][5];    // gm1, pr, pi, zr, zi
    __shared__ float s_coef[NSEC][5];   // b0, b1, b2, a1, a2
    __shared__ float s_mid[NSEC][9];    // g, rz, tz, zrp, zip, rp, tp, prp, pip
    __shared__ float s_grad[5][NSEC];   // dL/d{b0,b1,b2,a1,a2}

    const int tid  = threadIdx.x;
    const int lane = tid & 31;
    const int wv   = tid >> 5;
    const int sec  = lane & 15;   // section handled by this lane
    const int half = lane >> 4;   // 0: K=0,1 ; 1: K=2,3 of the WMMA A/B tiles

    // byte-index operands for ds_bpermute (lane^mask)*4, one per stage
    const int bp1 = ((lane ^ 1) << 2);
    const int bp2 = ((lane ^ 2) << 2);
    const int bp4 = ((lane ^ 4) << 2);
    const int bp8 = ((lane ^ 8) << 2);

    // ---- one-time init: trig tables, target, raw params ----
    __builtin_prefetch(target + tid, 0, 0);           // global_prefetch_b8
    {
        float w = PI_F * (float)tid / (float)(NF - 1);
        s_tgt[tid] = target[tid];
        s_c1[tid]  = __cosf(w);
        s_s1[tid]  = __sinf(w);
        s_c2[tid]  = __cosf(2.0f * w);
        s_s2[tid]  = __sinf(2.0f * w);
    }
    if (tid < NSEC) {
        #pragma unroll
        for (int j = 0; j < 5; ++j) s_raw[tid][j] = 1.0f;
    }
    __syncthreads();

    for (int iter = 0; iter < NIT; ++iter) {
        // ---- (a) forward coefficient build (16 threads) + grad zeroing ----
        if (tid < NSEC) {
            float gm1 = s_raw[tid][0], pr = s_raw[tid][1], pim = s_raw[tid][2];
            float zr  = s_raw[tid][3], zi = s_raw[tid][4];
            float g   = gm1 + 1.0f;
            float rz  = __builtin_sqrtf(zr * zr + zi * zi);
            float tz  = tanhf(rz);
            float sz  = tz / rz;
            float zrp = zr * sz, zip = zi * sz;
            float rp  = __builtin_sqrtf(pr * pr + pim * pim);
            float tp  = tanhf(rp);
            float sp  = tp / rp;
            float prp = pr * sp, pip = pim * sp;
            s_coef[tid][0] = g;
            s_coef[tid][1] = -2.0f * g * zrp;
            s_coef[tid][2] = g * (zrp * zrp + zip * zip);
            s_coef[tid][3] = -2.0f * prp;
            s_coef[tid][4] = prp * prp + pip * pip;
            s_mid[tid][0] = g;   s_mid[tid][1] = rz;  s_mid[tid][2] = tz;
            s_mid[tid][3] = zrp; s_mid[tid][4] = zip; s_mid[tid][5] = rp;
            s_mid[tid][6] = tp;  s_mid[tid][7] = prp; s_mid[tid][8] = pip;
        }
        if (tid >= 64 && tid < 64 + 5 * NSEC) {
            ((float*)s_grad)[tid - 64] = 0.0f;
        }
        __syncthreads();

        // outs[-1] in the reference is pz_to_sos of the state BEFORE the
        // 1000th update -> emit after 999 updates and stop.
        if (iter == NIT - 1) {
            if (tid < NSEC) {
                out[tid * 6 + 0] = s_coef[tid][0];
                out[tid * 6 + 1] = s_coef[tid][1];
                out[tid * 6 + 2] = s_coef[tid][2];
                out[tid * 6 + 3] = 1.0f;
                out[tid * 6 + 4] = s_coef[tid][3];
                out[tid * 6 + 5] = s_coef[tid][4];
            }
            break;  // uniform
        }

        const float b0 = s_coef[sec][0], b1 = s_coef[sec][1], b2 = s_coef[sec][2];
        const float a1 = s_coef[sec][3], a2 = s_coef[sec][4];

        // ---- (b1) pass 1: per-slot log2 |B/A|^2 for this lane's section ----
        float Ls[NSLOT];
        #pragma unroll
        for (int slot = 0; slot < NSLOT; ++slot) {
            const int grp = slot >> 1, k = slot & 1;
            const int f = wv * 32 + grp * 4 + half * 2 + k;
            const float c1 = s_c1[f], s1 = s_s1[f];
            const float c2 = s_c2[f], s2 = s_s2[f];
            const float reB = b0 + b1 * c1 + b2 * c2;
            const float imB = -(b1 * s1 + b2 * s2);
            const float reA = 1.0f + a1 * c1 + a2 * c2;
            const float imA = -(a1 * s1 + a2 * s2);
            const float mB  = reB * reB + imB * imB;
            const float mA  = reA * reA + imA * imA;
            Ls[slot] = __builtin_amdgcn_logf(mB) - __builtin_amdgcn_logf(mA);
        }

        // ---- (b2) batched butterfly: sum over the 16 sections per slot ----
        // 8 ds_bpermute per dscnt wait; 4 stages (xor 1,2,4,8 stay in-half)
        BFLY8(bp1, Ls, 0); BFLY8(bp1, Ls, 8);
        BFLY8(bp2, Ls, 0); BFLY8(bp2, Ls, 8);
        BFLY8(bp4, Ls, 0); BFLY8(bp4, Ls, 8);
        BFLY8(bp8, Ls, 0); BFLY8(bp8, Ls, 8);

        // ---- (b3) pass 2: weights + Jacobians + WMMA accumulation ----
        v8f Cb0 = {}, Cb1 = {}, Cb2 = {}, Ca1 = {}, Ca2 = {};
        #pragma unroll
        for (int grp = 0; grp < 8; ++grp) {
            v2f Ab0, Ab1, Ab2, Aa1, Aa2, Bw;
            #pragma unroll
            for (int k = 0; k < 2; ++k) {
                const int f = wv * 32 + grp * 4 + half * 2 + k;
                const float c1 = s_c1[f], s1 = s_s1[f];
                const float c2 = s_c2[f], s2 = s_s2[f];
                const float reB = b0 + b1 * c1 + b2 * c2;
                const float imB = -(b1 * s1 + b2 * s2);
                const float reA = 1.0f + a1 * c1 + a2 * c2;
                const float imA = -(a1 * s1 + a2 * s2);
                const float mB  = reB * reB + imB * imB;
                const float mA  = reA * reA + imA * imA;

                // |h| and dB from the section-summed log2 magnitude
                const float u   = __builtin_amdgcn_exp2f(0.5f * Ls[grp * 2 + k]);
                const float upe = u + EPSV;
                const float dB  = K20L2 * __builtin_amdgcn_logf(upe);
                const float wf  = (2.0f / (float)NF) * (dB - s_tgt[f]) *
                                  K20L * 0.5f * u * __builtin_amdgcn_rcpf(upe);

                const float iB = __builtin_amdgcn_rcpf(mB);
                const float iA = __builtin_amdgcn_rcpf(mA);
                Ab0[k] =  2.0f *  reB * iB;
                Ab1[k] =  2.0f * (reB * c1 - imB * s1) * iB;
                Ab2[k] =  2.0f * (reB * c2 - imB * s2) * iB;
                Aa1[k] = -2.0f * (reA * c1 - imA * s1) * iA;
                Aa2[k] = -2.0f * (reA * c2 - imA * s2) * iA;
                Bw[k]  = wf;    // broadcast across lanes == B-matrix rows
            }
            // D[sec, *] += sum_k J[sec, f_k] * w[f_k]  (full f32 WMMA)
            Cb0 = __builtin_amdgcn_wmma_f32_16x16x4_f32(false, Ab0, false, Bw, (short)0, Cb0, false, false);
            Cb1 = __builtin_amdgcn_wmma_f32_16x16x4_f32(false, Ab1, false, Bw, (short)0, Cb1, false, false);
            Cb2 = __builtin_amdgcn_wmma_f32_16x16x4_f32(false, Ab2, false, Bw, (short)0, Cb2, false, false);
            Ca1 = __builtin_amdgcn_wmma_f32_16x16x4_f32(false, Aa1, false, Bw, (short)0, Ca1, false, false);
            Ca2 = __builtin_amdgcn_wmma_f32_16x16x4_f32(false, Aa2, false, Bw, (short)0, Ca2, false, false);
        }

        // ---- (c) cross-wave combine: C tile rows -> LDS via ds_add_f32 ----
        // C layout: lanes 0-15 VGPR r = G[r], lanes 16-31 VGPR r = G[r+8];
        // columns are replicated, so one lane per half suffices.
        if (sec == 0) {
            const int ro = half * 8;
            #pragma unroll
            for (int r = 0; r < 8; ++r) {
                atomicAdd(&s_grad[0][ro + r], Cb0[r]);
                atomicAdd(&s_grad[1][ro + r], Cb1[r]);
                atomicAdd(&s_grad[2][ro + r], Cb2[r]);
                atomicAdd(&s_grad[3][ro + r], Ca1[r]);
                atomicAdd(&s_grad[4][ro + r], Ca2[r]);
            }
        }
        __syncthreads();

        // ---- (d) chain rule through tanh pole/zero map + SGD update ----
        if (tid < NSEC) {
            const float Gb0 = s_grad[0][tid], Gb1 = s_grad[1][tid];
            const float Gb2 = s_grad[2][tid];
            const float Ga1 = s_grad[3][tid], Ga2 = s_grad[4][tid];

            const float g   = s_mid[tid][0];
            const float rz  = s_mid[tid][1], tz = s_mid[tid][2];
            const float zrp = s_mid[tid][3], zip = s_mid[tid][4];
            const float rp  = s_mid[tid][5], tp = s_mid[tid][6];
            const float prp = s_mid[tid][7], pip = s_mid[tid][8];
            const float sz  = tz / rz;
            const float sp  = tp / rp;
            float gm1 = s_raw[tid][0], pr = s_raw[tid][1], pim = s_raw[tid][2];
            float zr  = s_raw[tid][3], zi = s_raw[tid][4];

            // grads w.r.t. scaled pole/zero coords and gain
            const float gg   = Gb0 - 2.0f * zrp * Gb1 + (zrp * zrp + zip * zip) * Gb2;
            const float gzrp = -2.0f * g * Gb1 + 2.0f * g * zrp * Gb2;
            const float gzip =  2.0f * g * zip * Gb2;
            const float gprp = -2.0f * Ga1 + 2.0f * prp * Ga2;
            const float gpip =  2.0f * pip * Ga2;

            // back through v' = v * tanh(|v|)/|v|:  g_v = s*g' + (s'/r)(v.g')v
            const float dsz  = (1.0f - tz * tz) / rz - tz / (rz * rz);
            const float dotz = zr * gzrp + zi * gzip;
            const float gzr  = sz * gzrp + (dsz / rz) * dotz * zr;
            const float gzi  = sz * gzip + (dsz / rz) * dotz * zi;
            const float dsp  = (1.0f - tp * tp) / rp - tp / (rp * rp);
            const float dotp = pr * gprp + pim * gpip;
            const float gpr  = sp * gprp + (dsp / rp) * dotp * pr;
            const float gpi  = sp * gpip + (dsp / rp) * dotp * pim;

            s_raw[tid][0] = gm1 - LRATE * gg;
            s_raw[tid][1] = pr  - LRATE * gpr;
            s_raw[tid][2] = pim - LRATE * gpi;
            s_raw[tid][3] = zr  - LRATE * gzr;
            s_raw[tid][4] = zi  - LRATE * gzi;
        }
        __syncthreads();
    }
}

extern "C" void kernel_launch(void* const* d_in, const int* in_sizes, int n_in,
                              void* d_out, int out_size, void* d_ws, size_t ws_size,
                              hipStream_t stream)
{
    (void)in_sizes; (void)n_in; (void)d_ws; (void)ws_size; (void)out_size;
    const float* target = (const float*)d_in[0];
    float*       out    = (float*)d_out;
    // One persistent workgroup: 512 threads = 16 wave32s on a single WGP.
    sgd_filter_kernel<<<1, 512, 0, stream>>>(target, out);
}